// SpatialRegularizationLoss_77738908057986
// MI455X (gfx1250) — compile-verified
//
#include <hip/hip_runtime.h>
#include <hip/hip_bf16.h>

// ---- CDNA5 WMMA types (wave32) ----
typedef __attribute__((ext_vector_type(2))) float v2f;
typedef __attribute__((ext_vector_type(8))) float v8f;

#define WAVES_PER_WG 8
#define WG_SIZE (WAVES_PER_WG * 32)
#define NPART 1024

// Single-instruction relu: v_med3_f32(x, 0, +inf) == clamp(x, 0, inf).
// (fmaxf(x,0) costs 2 VALU on gfx1250: canonicalize max(x,x) + max(0,x).)
__device__ __forceinline__ float relu1(float x) {
    return __builtin_amdgcn_fmed3f(x, 0.0f, __builtin_inff());
}

// One wave owns 16 edges; V_WMMA_F32_16X16X4_F32 computes the 16 per-edge dot
// products (the diagonal of A*B) in exact f32.
//
// Fragment layout (ISA 7.12.2, 32-bit operands), folded into the ADDRESS:
//   A 16x4: lanes 0-15 hold K={k,k+1} of src row, lanes 16-31 hold K={k+2,k+3}
//   B 4x16: lanes 0-15 hold K={k,k+1} of dst row, lanes 16-31 hold K={k+2,k+3}
// => lane L reads from rowBase + (L>=16 ? 2 : 0) + k, one b64 per operand per
//    step; lanes L and L+16 jointly cover a contiguous 16B chunk, so every
//    byte fetched is unique (no duplicate gathers, no shuffles, no selects).
//   C 16x16: diag n<8 at lane n in c[n]; diag n>=8 at lane n+16 in c[n-8].
//
// Tail edges (e >= E) need NO operand masking: they only influence A-row n /
// B-col n, which only reach diag entry (n,n), and that entry is skipped at
// extraction time. Control flow stays uniform through every WMMA (EXEC all 1s).
__device__ __forceinline__ void wmma_step4(const float* __restrict__ aRow,
                                           const float* __restrict__ bRow,
                                           int k, v8f& c) {
    float2 a2 = *(const float2*)(aRow + k);
    float2 b2 = *(const float2*)(bRow + k);
    v2f av, bv;
    av[0] = relu1(a2.x);  av[1] = relu1(a2.y);
    bv[0] = relu1(b2.x);  bv[1] = relu1(b2.y);
    c = __builtin_amdgcn_wmma_f32_16x16x4_f32(
            false, av, false, bv, (short)0, c, false, false);
}

template <bool ATOMIC_OUT>
__global__ void srl_tiles_kernel(const float* __restrict__ S,
                                 const float* __restrict__ pos,
                                 const int* __restrict__ ei,
                                 float* __restrict__ partials,  // NPART floats (or d_out)
                                 int K, int E, int nTiles, float scale) {
    const int lane = threadIdx.x & 31;
    const int wave = threadIdx.x >> 5;
    const int waveGlobal = blockIdx.x * WAVES_PER_WG + wave;
    const int waveStride = gridDim.x * WAVES_PER_WG;
    const bool hiHalf = (lane >= 16);
    const int hiOff = hiHalf ? 2 : 0;

    float acc = 0.0f;

    for (int t = waveGlobal; t < nTiles; t += waveStride) {
        const int n16 = lane & 15;
        const int e   = t * 16 + n16;
        const int ec  = (e < E) ? e : (E - 1);   // clamp: loads stay in-bounds
        const int si  = ei[ec];
        const int di  = ei[E + ec];

        const float* __restrict__ aRow = S + (size_t)si * (size_t)K + hiOff;
        const float* __restrict__ bRow = S + (size_t)di * (size_t)K + hiOff;

        v8f c = {0.f, 0.f, 0.f, 0.f, 0.f, 0.f, 0.f, 0.f};

        const int K8 = K & ~7;
        const int K4 = K & ~3;
        int k = 0;
        for (; k < K8; k += 8) {   // hot loop: 4 b64 gathers + 2 WMMAs
            wmma_step4(aRow, bRow, k,     c);
            wmma_step4(aRow, bRow, k + 4, c);
        }
        if (k < K4) {              // single 4-wide step (hit for K=100)
            wmma_step4(aRow, bRow, k, c);
            k += 4;
        }
        if (k < K) {               // K%4 tail, once, outside hot loop
            const int ko = k + hiOff;          // this lane's first K index
            v2f av, bv;
            av[0] = (ko < K)     ? relu1(aRow[k])     : 0.f;
            av[1] = (ko + 1 < K) ? relu1(aRow[k + 1]) : 0.f;
            bv[0] = (ko < K)     ? relu1(bRow[k])     : 0.f;
            bv[1] = (ko + 1 < K) ? relu1(bRow[k + 1]) : 0.f;
            c = __builtin_amdgcn_wmma_f32_16x16x4_f32(
                    false, av, false, bv, (short)0, c, false, false);
        }

        // Extract diagonal: lanes 0-7 -> diag[0..7] = c[lane];
        // lanes 24-31 -> diag[8..15] = c[lane-24].
        const bool has = (lane < 8) || (lane >= 24);
        if (has) {
            const int li = (lane < 8) ? lane : (lane - 24);
            float dv = c[0];
#pragma unroll
            for (int i = 1; i < 8; ++i) dv = (li == i) ? c[i] : dv;

            const int n  = (lane < 8) ? lane : (lane - 16);
            const int e2 = t * 16 + n;
            if (e2 < E) {
                const int s2 = ei[e2];
                const int d2 = ei[E + e2];
                const float* ps = pos + (size_t)s2 * 3u;
                const float* pd = pos + (size_t)d2 * 3u;
                const float dx = ps[0] - pd[0];
                const float dy = ps[1] - pd[1];
                const float dz = ps[2] - pd[2];
                acc += dv * (dx * dx + dy * dy + dz * dz);
            }
        }
    }

    // wave32 reduction
#pragma unroll
    for (int o = 16; o >= 1; o >>= 1) acc += __shfl_xor(acc, o, 32);

    __shared__ float sm[WAVES_PER_WG];
    if (lane == 0) sm[wave] = acc;
    __syncthreads();
    if (wave == 0) {
        float v = (lane < WAVES_PER_WG) ? sm[lane] : 0.f;
#pragma unroll
        for (int o = 4; o >= 1; o >>= 1) v += __shfl_xor(v, o, 32);
        if (lane == 0) {
            if (ATOMIC_OUT) {
                atomicAdd(partials, v * scale);
            } else {
                partials[blockIdx.x] = v;
            }
        }
    }
}

__global__ void srl_zero_kernel(float* out) { out[0] = 0.0f; }

__global__ void srl_reduce_kernel(const float* __restrict__ partials, int n,
                                  float scale, float* __restrict__ out) {
    const int lane = threadIdx.x & 31;
    const int wave = threadIdx.x >> 5;
    float v = 0.f;
    for (int i = threadIdx.x; i < n; i += WG_SIZE) v += partials[i];
#pragma unroll
    for (int o = 16; o >= 1; o >>= 1) v += __shfl_xor(v, o, 32);
    __shared__ float sm[WAVES_PER_WG];
    if (lane == 0) sm[wave] = v;
    __syncthreads();
    if (threadIdx.x == 0) {
        float s = 0.f;
        for (int i = 0; i < WAVES_PER_WG; ++i) s += sm[i];
        out[0] = s * scale;
    }
}

extern "C" void kernel_launch(void* const* d_in, const int* in_sizes, int n_in,
                              void* d_out, int out_size, void* d_ws, size_t ws_size,
                              hipStream_t stream) {
    const float* S   = (const float*)d_in[0];   // [N, K] f32
    const float* pos = (const float*)d_in[1];   // [N, 3] f32
    const int*   ei  = (const int*)d_in[2];     // [2, E] i32
    float* out = (float*)d_out;

    const int P = 3;
    const int N = in_sizes[1] / P;
    const int K = in_sizes[0] / N;
    const int E = in_sizes[2] / 2;
    (void)N;

    const int nTiles = (E + 15) / 16;
    const float scale = 0.01f / (float)E;   // WEIGHT / num_edges

    if (ws_size >= (size_t)NPART * sizeof(float)) {
        float* partials = (float*)d_ws;
        // grid == NPART so every partial slot is (re)written each call
        srl_tiles_kernel<false><<<NPART, WG_SIZE, 0, stream>>>(
            S, pos, ei, partials, K, E, nTiles, scale);
        srl_reduce_kernel<<<1, WG_SIZE, 0, stream>>>(partials, NPART, scale, out);
    } else {
        // fallback: zero + atomic accumulation directly into d_out
        srl_zero_kernel<<<1, 1, 0, stream>>>(out);
        srl_tiles_kernel<true><<<NPART, WG_SIZE, 0, stream>>>(
            S, pos, ei, out, K, E, nTiles, scale);
    }
}